// MostSimilarImage_30580167147970
// MI455X (gfx1250) — compile-verified
//
#include <hip/hip_runtime.h>

typedef __attribute__((ext_vector_type(2))) float v2f;
typedef __attribute__((ext_vector_type(4))) float v4f;
typedef __attribute__((ext_vector_type(8))) float v8f;

#define B_  32
#define C_  512
#define HW_ 64
#define N_  400000
#define L_  100
#define T_  80
#define V_  5000
#define ONEHOT_ (B_*T_*V_)   /* 12,800,000 floats */

// ---------------------------------------------------------------------------
// 1) features[b][c] = max over the 64 contiguous HW elements
// ---------------------------------------------------------------------------
__global__ void k_features(const float* __restrict__ fmaps, float* __restrict__ feat) {
    int t = blockIdx.x * blockDim.x + threadIdx.x;   // 16384 = 32*512
    const float4* p = (const float4*)(fmaps + (size_t)t * HW_);
    float m = -3.402823466e38f;
#pragma unroll
    for (int i = 0; i < 16; ++i) {
        float4 v = p[i];
        m = fmaxf(m, fmaxf(fmaxf(v.x, v.y), fmaxf(v.z, v.w)));
    }
    feat[t] = m;
}

// ---------------------------------------------------------------------------
// 2) x2[b] = sum_c features[b][c]^2
// ---------------------------------------------------------------------------
__global__ void k_x2(const float* __restrict__ feat, float* __restrict__ x2) {
    __shared__ float sm[64];
    int b = blockIdx.x, t = threadIdx.x;
    float s = 0.f;
    for (int c = t; c < C_; c += 64) { float v = feat[b * C_ + c]; s += v * v; }
    sm[t] = s; __syncthreads();
    for (int off = 32; off > 0; off >>= 1) {
        if (t < off) sm[t] += sm[t + off];
        __syncthreads();
    }
    if (t == 0) x2[b] = sm[0];
}

// ---------------------------------------------------------------------------
// 3) GEMM (WMMA f32 16x16x4) + fused y2 + distance epilogue.
//    One wave: 16 columns of N, both M-tiles (M=0..15 and 16..31).
//    K order is permuted so every lane loads 16B contiguous (b128):
//      hi=0 lane loads k=8j..8j+3, hi=1 lane loads k=8j+4..8j+7.
//      WMMA#1 uses .xy halves, WMMA#2 uses .zw halves; A is addressed from
//      LDS with the same hi*4 skew, so A/B k-slots match lane-for-lane.
// ---------------------------------------------------------------------------
__global__ void __launch_bounds__(256) k_gemm(const float* __restrict__ allf,
                                              const float* __restrict__ feat,
                                              const float* __restrict__ x2,
                                              float* __restrict__ dist) {
    __shared__ __align__(16) float ldsA[B_ * C_];    // 64 KB: full A matrix

    int tid = threadIdx.x;
    // stage features (32x512 f32) into LDS: 4096 float4 / 256 threads
    {
        const float4* src = (const float4*)feat;
        float4* dst = (float4*)ldsA;
#pragma unroll
        for (int i = 0; i < 16; ++i) dst[tid + i * 256] = src[tid + i * 256];
    }
    __syncthreads();

    int w    = tid >> 5;          // wave in block (0..7)
    int lane = tid & 31;
    int nl   = lane & 15;
    int hi   = lane >> 4;         // k-slot skew: 0 -> +0, 1 -> +4 floats
    int tile_n = (blockIdx.x * 8 + w) * 16;
    int n = tile_n + nl;

    const v4f* gb  = (const v4f*)(allf + (size_t)n * C_ + hi * 4);
    const v4f* la0 = (const v4f*)(ldsA + nl * C_ + hi * 4);
    const v4f* la1 = (const v4f*)(ldsA + (nl + 16) * C_ + hi * 4);

    v8f acc0 = {};   // M = 0..15
    v8f acc1 = {};   // M = 16..31
    float y2p = 0.f;

#pragma unroll 2
    for (int j = 0; j < C_ / 8; ++j) {
        v4f bq  = gb[2 * j];     // 8 k's per iter: this lane's contiguous 4
        v4f a0q = la0[2 * j];
        v4f a1q = la1[2 * j];
        v2f blo = bq.xy,  bhi = bq.zw;
        v2f a0lo = a0q.xy, a0hi = a0q.zw;
        v2f a1lo = a1q.xy, a1hi = a1q.zw;
        acc0 = __builtin_amdgcn_wmma_f32_16x16x4_f32(false, a0lo, false, blo,
                                                     (short)0, acc0, false, false);
        acc1 = __builtin_amdgcn_wmma_f32_16x16x4_f32(false, a1lo, false, blo,
                                                     (short)0, acc1, false, false);
        acc0 = __builtin_amdgcn_wmma_f32_16x16x4_f32(false, a0hi, false, bhi,
                                                     (short)0, acc0, false, false);
        acc1 = __builtin_amdgcn_wmma_f32_16x16x4_f32(false, a1hi, false, bhi,
                                                     (short)0, acc1, false, false);
        y2p += bq.x * bq.x + bq.y * bq.y + bq.z * bq.z + bq.w * bq.w;
    }

    // lanes l and l+16 each hold half of ||row n||^2
    float y2 = y2p + __shfl_xor(y2p, 16, 32);

    // C/D layout: VGPR r -> M=r (lanes 0..15, N=lane) / M=r+8 (lanes 16..31, N=lane-16)
#pragma unroll
    for (int r = 0; r < 8; ++r) {
        int m0 = hi * 8 + r;
        float d0 = x2[m0] + y2 - 2.0f * acc0[r];
        dist[(size_t)m0 * N_ + n] = sqrtf(fmaxf(d0, 0.0f));
        int m1 = 16 + m0;
        float d1 = x2[m1] + y2 - 2.0f * acc1[r];
        dist[(size_t)m1 * N_ + n] = sqrtf(fmaxf(d1, 0.0f));
    }
}

// ---------------------------------------------------------------------------
// 4) argmin over each of the 32 rows (first occurrence on ties, like jnp.argmin)
// ---------------------------------------------------------------------------
__global__ void k_argmin(const float* __restrict__ dist, int* __restrict__ closest) {
    __shared__ float sv[256];
    __shared__ int   si[256];
    int m = blockIdx.x, t = threadIdx.x;
    const float* row = dist + (size_t)m * N_;
    float best = 3.402823466e38f; int bi = N_;
    for (int i = t; i < N_; i += 256) {
        float v = row[i];
        if (v < best) { best = v; bi = i; }   // i increasing -> keeps first occurrence
    }
    sv[t] = best; si[t] = bi; __syncthreads();
    for (int off = 128; off > 0; off >>= 1) {
        if (t < off) {
            float v = sv[t + off]; int i2 = si[t + off];
            if (v < sv[t] || (v == sv[t] && i2 < si[t])) { sv[t] = v; si[t] = i2; }
        }
        __syncthreads();
    }
    if (t == 0) closest[m] = si[0];
}

// ---------------------------------------------------------------------------
// 5) one-hot: zero then scatter ones
// ---------------------------------------------------------------------------
__global__ void k_zero(float4* __restrict__ out) {
    out[(size_t)blockIdx.x * 256 + threadIdx.x] = make_float4(0.f, 0.f, 0.f, 0.f);
}

__global__ void k_scatter(const long long* __restrict__ allrep,
                          const int* __restrict__ closest,
                          float* __restrict__ oh) {
    int id = blockIdx.x * blockDim.x + threadIdx.x;   // 2560 = 32*80
    int b = id / T_, t = id % T_;
    long long v = allrep[(size_t)closest[b] * L_ + t];
    if (v >= 0 && v < V_)
        oh[(size_t)b * T_ * V_ + (size_t)t * V_ + (int)v] = 1.0f;
}

// ---------------------------------------------------------------------------
extern "C" void kernel_launch(void* const* d_in, const int* in_sizes, int n_in,
                              void* d_out, int out_size, void* d_ws, size_t ws_size,
                              hipStream_t stream) {
    const float*     fmaps  = (const float*)d_in[0];       // (32,512,8,8) f32
    const float*     allf   = (const float*)d_in[1];       // (400000,512) f32
    const long long* allrep = (const long long*)d_in[2];   // (400000,100) i64
    // d_in[3] = reports (shape only), d_in[4] = vocab_size (constant 5000)

    float* out = (float*)d_out;                // [one_hot (12.8M) | distances (12.8M)]
    char*  ws  = (char*)d_ws;
    float* feat    = (float*)ws;               // 64 KB
    float* x2      = (float*)(ws + 65536);     // 128 B
    int*   closest = (int*)(ws + 65536 + 128); // 128 B
    float* dist    = out + ONEHOT_;

    k_features<<<64,   256, 0, stream>>>(fmaps, feat);
    k_x2      <<<32,    64, 0, stream>>>(feat, x2);
    k_gemm    <<<3125, 256, 0, stream>>>(allf, feat, x2, dist);
    k_argmin  <<<32,   256, 0, stream>>>(dist, closest);
    k_zero    <<<12500,256, 0, stream>>>((float4*)out);
    k_scatter <<<10,   256, 0, stream>>>(allrep, closest, out);
}